// RecyclingEmbedder_36395552866967
// MI455X (gfx1250) — compile-verified
//
#include <hip/hip_runtime.h>
#include <stdint.h>

// ---------------------------------------------------------------------------
// RecyclingEmbedder for MI455X (gfx1250, wave32).
// Memory-bound streaming: ~604MB traffic -> ~26us roofline @ 23.3TB/s.
// Pass 1: per-pair bin indices (lane-parallel, u8 table in d_ws).
// Pass 2: CDNA5 async global->LDS (ASYNCcnt) pipeline, 8-deep per wave
//         (4KB reads in flight/wave), b128, NT stores, wave32 reductions.
// ---------------------------------------------------------------------------

typedef float v4f __attribute__((ext_vector_type(4)));
typedef int   v4i __attribute__((ext_vector_type(4)));

#define PAIR_DIM 128
#define MSA_DIM  256
#define NBINS    15
#define LN_EPS   1e-5f

#define CHUNK          8                       // rows per async chunk
#define NCHUNK         4                       // chunks per wave
#define ROWS_PER_WAVE  (CHUNK * NCHUNK)        // 32
#define ROWS_PER_BLOCK (8 * ROWS_PER_WAVE)     // 256 (8 waves)
#define WT_BYTES       (NBINS * PAIR_DIM * 4)  // 7680

#if defined(__HIP_DEVICE_COMPILE__) && __has_builtin(__builtin_amdgcn_global_load_async_to_lds_b128)
#define USE_ASYNC_LDS 1
#else
#define USE_ASYNC_LDS 0
#endif

typedef __attribute__((address_space(1))) v4i gv4i;  // global int4
typedef __attribute__((address_space(3))) v4i lv4i;  // LDS int4

__device__ __forceinline__ void wait_async0() {
#if defined(__HIP_DEVICE_COMPILE__) && __has_builtin(__builtin_amdgcn_s_wait_asynccnt)
  __builtin_amdgcn_s_wait_asynccnt(0);
#else
  asm volatile("s_wait_asynccnt 0" ::: "memory");
#endif
  asm volatile("" ::: "memory");
}
__device__ __forceinline__ void wait_async8() {
#if defined(__HIP_DEVICE_COMPILE__) && __has_builtin(__builtin_amdgcn_s_wait_asynccnt)
  __builtin_amdgcn_s_wait_asynccnt(8);
#else
  asm volatile("s_wait_asynccnt 8" ::: "memory");
#endif
  asm volatile("" ::: "memory");
}
__device__ __forceinline__ void wait_ds0() {
#if defined(__HIP_DEVICE_COMPILE__) && __has_builtin(__builtin_amdgcn_s_wait_dscnt)
  __builtin_amdgcn_s_wait_dscnt(0);
#else
  asm volatile("s_wait_dscnt 0" ::: "memory");
#endif
  asm volatile("" ::: "memory");
}

__device__ __forceinline__ float wave_reduce_add(float v) {
#pragma unroll
  for (int off = 16; off > 0; off >>= 1) v += __shfl_xor(v, off, 32);
  return v;
}

// Bins: concat([3.375], arange(5.125, 22, 1.25)). Sorted => nearest-bin
// argmin == count of midpoints strictly below d. Midpoints are exact in f32.
// Strict '>' matches jnp.argmin's first-minimum tie rule (tie -> lower bin).
__device__ __forceinline__ int bin_index(float d) {
  const float mids[NBINS - 1] = {4.25f,  5.75f,  7.0f,  8.25f, 9.5f,
                                 10.75f, 12.0f,  13.25f, 14.5f, 15.75f,
                                 17.0f,  18.25f, 19.5f,  20.75f};
  int idx = 0;
#pragma unroll
  for (int k = 0; k < NBINS - 1; ++k) idx += (d > mids[k]) ? 1 : 0;
  return idx;
}

__device__ __forceinline__ int bin_from_ij(int i, int j,
                                           const float* __restrict__ ca) {
  float dx = ca[3 * i + 0] - ca[3 * j + 0];
  float dy = ca[3 * i + 1] - ca[3 * j + 1];
  float dz = ca[3 * i + 2] - ca[3 * j + 2];
  return bin_index(sqrtf(dx * dx + dy * dy + dz * dz));
}

// Pass 1: one thread per pair -> u8 bin index table (lane-parallel).
__global__ __launch_bounds__(256) void recyc_bins_kernel(
    const float* __restrict__ ca, unsigned char* __restrict__ idxbuf,
    int N, long long nrows) {
  long long r = (long long)blockIdx.x * blockDim.x + threadIdx.x;
  if (r >= nrows) return;
  int i = (int)(r / (long long)N);
  int j = (int)(r - (long long)i * (long long)N);
  idxbuf[r] = (unsigned char)bin_from_ij(i, j, ca);
}

// wT[bin][c] = w_onehot[c, bin] + b_onehot[c]   (15 x 128 floats in d_ws)
__global__ void recyc_prep_kernel(const float* __restrict__ w,
                                  const float* __restrict__ b,
                                  float* __restrict__ wT) {
  for (int t = threadIdx.x; t < NBINS * PAIR_DIM; t += blockDim.x) {
    int k = t >> 7;       // bin
    int ch = t & 127;     // channel
    wT[t] = w[ch * NBINS + k] + b[ch];
  }
}

// One pair row: LN + dist_emb. x = this lane's 4 channels, idx = bin.
__device__ __forceinline__ void pair_process(
    v4f x, long long r, int idx, int lane,
    const float* __restrict__ w, const float* __restrict__ wT,
    v4f g4, v4f be4, v4f b4, float* __restrict__ out) {
  float s = x.x + x.y + x.z + x.w;
  float q = x.x * x.x + x.y * x.y + x.z * x.z + x.w * x.w;
  s = wave_reduce_add(s);
  q = wave_reduce_add(q);
  const float invD = 1.0f / (float)PAIR_DIM;
  float mean = s * invD;
  float var  = q * invD - mean * mean;
  float inv  = rsqrtf(var + LN_EPS);

  int c = lane * 4;
  v4f e;  // dist_emb with bias folded in
  if (wT) {
    e = *(const v4f*)(wT + idx * PAIR_DIM + c);
  } else {
    e.x = w[(c + 0) * NBINS + idx] + b4.x;
    e.y = w[(c + 1) * NBINS + idx] + b4.y;
    e.z = w[(c + 2) * NBINS + idx] + b4.z;
    e.w = w[(c + 3) * NBINS + idx] + b4.w;
  }

  v4f o;
  o.x = (x.x - mean) * inv * g4.x + be4.x + e.x;
  o.y = (x.y - mean) * inv * g4.y + be4.y + e.y;
  o.z = (x.z - mean) * inv * g4.z + be4.z + e.z;
  o.w = (x.w - mean) * inv * g4.w + be4.w + e.w;
  __builtin_nontemporal_store(o, (v4f*)(out + r * (long long)PAIR_DIM + c));
}

// 256 threads = 8 waves; each wave streams 32 contiguous rows as 4 chunks of
// 8 rows, double-buffered through LDS via async loads (8 in flight).
__global__ __launch_bounds__(256) void recyc_pair_kernel(
    const float* __restrict__ pair_rep, const float* __restrict__ ca,
    const float* __restrict__ w_onehot, const float* __restrict__ b_onehot,
    const float* __restrict__ gamma, const float* __restrict__ beta,
    const float* __restrict__ wT, const unsigned char* __restrict__ idxbuf,
    float* __restrict__ out, int N, long long nrows) {
  const int lane = threadIdx.x & 31;
  const int wave = threadIdx.x >> 5;
  const int c = lane * 4;

  v4f g4  = *(const v4f*)(gamma + c);
  v4f be4 = *(const v4f*)(beta + c);
  v4f b4  = *(const v4f*)(b_onehot + c);

  const long long wbase =
      (long long)blockIdx.x * ROWS_PER_BLOCK + (long long)wave * ROWS_PER_WAVE;

  // Fallback-only incremental (i,j): one div per wave, wave-uniform updates.
  int i = (int)(wbase / (long long)N);
  int j = (int)(wbase - (long long)i * (long long)N);

#if USE_ASYNC_LDS
  __shared__ __align__(16) float lbuf[2][8][CHUNK * PAIR_DIM];  // 64KB

  #define CHUNK_FULL(cc) (wbase + ((cc) + 1ll) * CHUNK <= nrows)
  #define ISSUE_CHUNK(cc, buf)                                                \
    do {                                                                      \
      _Pragma("unroll") for (int k = 0; k < CHUNK; ++k) {                     \
        long long rr = wbase + (long long)(cc) * CHUNK + k;                   \
        __builtin_amdgcn_global_load_async_to_lds_b128(                       \
            (gv4i*)(uintptr_t)(pair_rep + rr * (long long)PAIR_DIM + c),      \
            (lv4i*)(&lbuf[(buf)][wave][k * PAIR_DIM + c]), 0, 0);             \
      }                                                                       \
    } while (0)

  if (CHUNK_FULL(0)) ISSUE_CHUNK(0, 0);

  for (int cc = 0; cc < NCHUNK; ++cc) {
    bool curFull = CHUNK_FULL(cc);
    bool nxtFull = (cc + 1 < NCHUNK) && CHUNK_FULL(cc + 1);
    if (nxtFull) {
      wait_ds0();  // prior reads of the buffer being refilled are retired
      ISSUE_CHUNK(cc + 1, (cc + 1) & 1);
      wait_async8();  // in-order completion => current chunk resident
    } else if (curFull) {
      wait_async0();
    }
    if (curFull) {
      // 8 bin indices in one u64 (wbase+cc*8 is 8B aligned; buffer padded).
      uint64_t packed = 0;
      if (idxbuf)
        packed = *(const uint64_t*)(idxbuf + wbase + (long long)cc * CHUNK);
#pragma unroll
      for (int k = 0; k < CHUNK; ++k) {
        long long r = wbase + (long long)cc * CHUNK + k;
        int idx = idxbuf ? (int)((packed >> (8 * k)) & 0xff)
                         : bin_from_ij(i, j, ca);
        v4f x = *(const v4f*)(&lbuf[cc & 1][wave][k * PAIR_DIM + c]);
        pair_process(x, r, idx, lane, w_onehot, wT, g4, be4, b4, out);
        if (++j == N) { j = 0; ++i; }
      }
    } else {
      // tail: partial chunk, direct loads
      for (int k = 0; k < CHUNK; ++k) {
        long long r = wbase + (long long)cc * CHUNK + k;
        if (r < nrows) {
          int idx = idxbuf ? (int)idxbuf[r] : bin_from_ij(i, j, ca);
          v4f x = __builtin_nontemporal_load(
              (const v4f*)(pair_rep + r * (long long)PAIR_DIM + c));
          pair_process(x, r, idx, lane, w_onehot, wT, g4, be4, b4, out);
        }
        if (++j == N) { j = 0; ++i; }
      }
    }
  }
  #undef ISSUE_CHUNK
  #undef CHUNK_FULL
#else
  for (int t = 0; t < ROWS_PER_WAVE; ++t) {
    long long r = wbase + t;
    if (r < nrows) {
      int idx = idxbuf ? (int)idxbuf[r] : bin_from_ij(i, j, ca);
      v4f x = __builtin_nontemporal_load(
          (const v4f*)(pair_rep + r * (long long)PAIR_DIM + c));
      pair_process(x, r, idx, lane, w_onehot, wT, g4, be4, b4, out);
    }
    if (++j == N) { j = 0; ++i; }
  }
#endif
}

// LN over 256 channels: one wave per row, 8 floats (2 x b128) per lane.
__global__ __launch_bounds__(256) void recyc_msa_kernel(
    const float* __restrict__ msa, const float* __restrict__ gamma,
    const float* __restrict__ beta, float* __restrict__ out, int N) {
  int row  = (int)((blockIdx.x * blockDim.x + threadIdx.x) >> 5);
  int lane = threadIdx.x & 31;
  if (row >= N) return;
  const float* rp = msa + (size_t)row * MSA_DIM;
  int c0 = lane * 4, c1 = 128 + lane * 4;
  v4f a = __builtin_nontemporal_load((const v4f*)(rp + c0));
  v4f b = __builtin_nontemporal_load((const v4f*)(rp + c1));
  float s = a.x + a.y + a.z + a.w + b.x + b.y + b.z + b.w;
  float q = a.x * a.x + a.y * a.y + a.z * a.z + a.w * a.w +
            b.x * b.x + b.y * b.y + b.z * b.z + b.w * b.w;
  s = wave_reduce_add(s);
  q = wave_reduce_add(q);
  const float invD = 1.0f / (float)MSA_DIM;
  float mean = s * invD;
  float var  = q * invD - mean * mean;
  float inv  = rsqrtf(var + LN_EPS);
  v4f g0  = *(const v4f*)(gamma + c0), g1  = *(const v4f*)(gamma + c1);
  v4f be0 = *(const v4f*)(beta + c0),  be1 = *(const v4f*)(beta + c1);
  v4f o0, o1;
  o0.x = (a.x - mean) * inv * g0.x + be0.x;
  o0.y = (a.y - mean) * inv * g0.y + be0.y;
  o0.z = (a.z - mean) * inv * g0.z + be0.z;
  o0.w = (a.w - mean) * inv * g0.w + be0.w;
  o1.x = (b.x - mean) * inv * g1.x + be1.x;
  o1.y = (b.y - mean) * inv * g1.y + be1.y;
  o1.z = (b.z - mean) * inv * g1.z + be1.z;
  o1.w = (b.w - mean) * inv * g1.w + be1.w;
  float* op = out + (size_t)row * MSA_DIM;
  __builtin_nontemporal_store(o0, (v4f*)(op + c0));
  __builtin_nontemporal_store(o1, (v4f*)(op + c1));
}

extern "C" void kernel_launch(void* const* d_in, const int* in_sizes, int n_in,
                              void* d_out, int out_size, void* d_ws, size_t ws_size,
                              hipStream_t stream) {
  const float* msa_row  = (const float*)d_in[0];  // [1,N,256]
  const float* pair_rep = (const float*)d_in[1];  // [1,N,N,128]
  const float* ca       = (const float*)d_in[2];  // [1,N,3]
  const float* w_onehot = (const float*)d_in[3];  // [128,15]
  const float* b_onehot = (const float*)d_in[4];  // [128]
  const float* g_pair   = (const float*)d_in[5];
  const float* b_pair   = (const float*)d_in[6];
  const float* g_msa    = (const float*)d_in[7];
  const float* b_msa    = (const float*)d_in[8];

  const int N = in_sizes[2] / 3;
  const long long nrows = (long long)N * (long long)N;

  float* out      = (float*)d_out;
  float* msa_out  = out;                        // N*256 floats
  float* pair_out = out + (size_t)N * MSA_DIM;  // N*N*128 floats

  // Workspace: [wT: 7680 B][idxbuf: nrows + 8 B]
  float* wT = nullptr;
  unsigned char* idxbuf = nullptr;
  if (ws_size >= (size_t)WT_BYTES) {
    wT = (float*)d_ws;
    recyc_prep_kernel<<<1, 256, 0, stream>>>(w_onehot, b_onehot, wT);
    if (ws_size >= (size_t)WT_BYTES + (size_t)nrows + 8) {
      idxbuf = (unsigned char*)d_ws + WT_BYTES;
      long long bin_blocks = (nrows + 255) / 256;
      recyc_bins_kernel<<<dim3((unsigned)bin_blocks), 256, 0, stream>>>(
          ca, idxbuf, N, nrows);
    }
  }

  int msa_blocks = (N + 7) / 8;  // 8 waves/block, 1 wave/row
  recyc_msa_kernel<<<msa_blocks, 256, 0, stream>>>(msa_row, g_msa, b_msa,
                                                   msa_out, N);

  long long pair_blocks = (nrows + ROWS_PER_BLOCK - 1) / ROWS_PER_BLOCK;
  recyc_pair_kernel<<<dim3((unsigned)pair_blocks), 256, 0, stream>>>(
      pair_rep, ca, w_onehot, b_onehot, g_pair, b_pair, wT, idxbuf,
      pair_out, N, nrows);
}